// DawsonIntegrate_22505628631234
// MI455X (gfx1250) — compile-verified
//
#include <hip/hip_runtime.h>
#include <cstdint>

#define BLOCK 256

// -0.25 * EulerGamma
#define NEG_QGAMMA (-0.14430391622538322f)
#define SQRT_PI    (1.7724539f)

// ---------- asymptotic branch: |x| > 6 (rare: ~6e-5 of lanes) ----------
__device__ __forceinline__ float asym_far(float ax) {
    // v = -ax ; y = -0.25*g - 0.5*log(-2v) + inv2*(-0.125 + inv2*(3/32 - inv2*5/32))
    float inv2  = 1.0f / (ax * ax);
    float inner = fmaf(inv2, -0.15625f, 0.09375f);
    float mid   = fmaf(inv2, inner, -0.125f);
    float base  = NEG_QGAMMA - 0.5f * logf(2.0f * ax);
    return fmaf(inv2, mid, base);
}

// ---------- Dawson region C (x >= 6.25), only reached via the rare far guard ----------
__device__ __forceinline__ float temp_far_C(float xe) {
    float t = xe * xe;
    float w = __builtin_amdgcn_rcpf(t);
    float pc = -5.90592860534773254987e-1f;
    pc = fmaf(pc, w, 6.29235242724368800674e-1f);
    pc = fmaf(pc, w, -1.72858975380388136411e-1f);
    pc = fmaf(pc, w, 1.64837047825189632310e-2f);
    pc = fmaf(pc, w, -4.86827613020462700845e-4f);
    float qc = w + (-2.69820057197544900361f);
    qc = fmaf(qc, w, 1.73270799045947845857f);
    qc = fmaf(qc, w, -3.93708582562939564892e-1f);
    qc = fmaf(qc, w, 3.44278924041233391079e-2f);
    qc = fmaf(qc, w, -9.73655226040941223894e-4f);
    float numC = fmaf(w, pc, qc);
    float denC = 2.0f * xe * qc;
    float F = numC * __builtin_amdgcn_rcpf(denC);
    return SQRT_PI * expf(t) * F;
}

// ---------- per-element hot path: cheb piecewise + sqrt(pi)*exp(x^2)*Dawson(x) ----------
__device__ __forceinline__ void eval_core(float xi, const float* __restrict__ sC,
                                          float& ynear, float& temp) {
    float ax = fabsf(xi);

    // Chebyshev piecewise (Clenshaw), v in [-6, 0]; safe v=-3 for far lanes
    float v = (ax > 6.0f) ? -3.0f : -ax;
    int b = (int)ceilf((v + 6.0f) / 1.5f) - 1;     // exact same fp32 ops as reference
    b = b < 0 ? 0 : (b > 3 ? 3 : b);
    const float* C = sC + (b << 3);                // LDS gathers (ds_load_b128 pairs)
    float c0 = C[6];
    float c1 = C[7];
    float tv = 2.0f * v;
#pragma unroll
    for (int k = 5; k >= 0; --k) {
        float n0 = C[k] - c1;
        c1 = fmaf(c1, tv, c0);
        c0 = n0;
    }
    ynear = fmaf(c1, v, c0);

    // pi/2*erfi(max(x,0)) = sqrt(pi)*exp(x^2)*Dawson(x); Cephes dawsn, regions A/B
    float xe = xi > 0.0f ? xi : 0.0f;
    float t  = xe * xe;

    // region A: x < 3.25, rational in t
    float pa = 1.13681498971755972054e-11f;
    pa = fmaf(pa, t, 8.49262267667473811108e-10f);
    pa = fmaf(pa, t, 1.94434204175553054283e-8f);
    pa = fmaf(pa, t, 9.53151741254484363489e-7f);
    pa = fmaf(pa, t, 3.07828309874913200438e-6f);
    pa = fmaf(pa, t, 3.52513368520288738649e-4f);
    pa = fmaf(pa, t, -8.50149846724410912031e-4f);
    pa = fmaf(pa, t, 4.22618223005546594270e-2f);
    pa = fmaf(pa, t, -9.17480371773452345351e-2f);
    pa = fmaf(pa, t, 1.0f);
    float qa = 2.40372073066762605484e-11f;
    qa = fmaf(qa, t, 1.48864681368493396752e-9f);
    qa = fmaf(qa, t, 5.21265281010541664570e-8f);
    qa = fmaf(qa, t, 1.27258478273186970203e-6f);
    qa = fmaf(qa, t, 2.32490249820789513991e-5f);
    qa = fmaf(qa, t, 3.25524741826057911661e-4f);
    qa = fmaf(qa, t, 3.48805814657162590916e-3f);
    qa = fmaf(qa, t, 2.79448531198828973716e-2f);
    qa = fmaf(qa, t, 1.58874241960120565368e-1f);
    qa = fmaf(qa, t, 5.74918629489320327824e-1f);
    qa = fmaf(qa, t, 1.0f);
    float numA = xe * pa, denA = qa;

    // region B: 3.25 <= x (valid to 6.25; lanes beyond get fixed up in far guard)
    float w = __builtin_amdgcn_rcpf(fmaxf(t, 1.0e-20f));
    float pb = 5.08955156417900903354e-1f;
    pb = fmaf(pb, w, -2.44754418142697847934e-1f);
    pb = fmaf(pb, w, 9.41512335303534411857e-2f);
    pb = fmaf(pb, w, -2.18711255142039025206e-2f);
    pb = fmaf(pb, w, 3.66207612329569181322e-3f);
    pb = fmaf(pb, w, -4.23209114460388756528e-4f);
    pb = fmaf(pb, w, 3.59641304793896631888e-5f);
    pb = fmaf(pb, w, -2.14640351719968974225e-6f);
    pb = fmaf(pb, w, 9.10010780076391431042e-8f);
    pb = fmaf(pb, w, -2.40274520828250956942e-9f);
    pb = fmaf(pb, w, 3.59233385440928410398e-11f);
    float qb = w + (-6.31839869873368190192e-1f);
    qb = fmaf(qb, w, 2.36706788228248691528e-1f);
    qb = fmaf(qb, w, -5.31806367003999806931e-2f);
    qb = fmaf(qb, w, 8.48041718586295374409e-3f);
    qb = fmaf(qb, w, -9.47996768486665330168e-4f);
    qb = fmaf(qb, w, 7.81025592944552338085e-5f);
    qb = fmaf(qb, w, -4.55875153252442634831e-6f);
    qb = fmaf(qb, w, 1.89100358111421846170e-7f);
    qb = fmaf(qb, w, -4.91324691331920606875e-9f);
    qb = fmaf(qb, w, 7.18466403235734541950e-11f);
    float numB = fmaf(w, pb, qb);
    float denB = 2.0f * xe * qb;

    bool rB = xe >= 3.25f;
    float num = rB ? numB : numA;
    float den = rB ? denB : denA;
    float F = num * __builtin_amdgcn_rcpf(den);   // single reciprocal serves both

    temp = SQRT_PI * expf(t) * F;                  // 0 for non-positive lanes (F(0)=0)
}

__global__ __launch_bounds__(BLOCK) void dawson_g_kernel(
    const float* __restrict__ x, const float* __restrict__ cheb,
    float* __restrict__ out, int n4, int rem) {
    // Stage the 32-float Chebyshev table into LDS with the gfx1250 async-to-LDS
    // path. Every lane stages sC[tid&31] (waves duplicate-write identical data),
    // so there is no EXEC-diverged region and every wave uniformly executes the
    // s_wait_asynccnt before the barrier.
    __shared__ float sC[32];
    {
        unsigned idx = threadIdx.x & 31u;
        unsigned lds_off = (unsigned)(uintptr_t)(&sC[idx]);
        unsigned long long gaddr = (unsigned long long)(uintptr_t)(cheb + idx);
        asm volatile("global_load_async_to_lds_b32 %0, %1, off"
                     :: "v"(lds_off), "v"(gaddr) : "memory");
        asm volatile("s_wait_asynccnt 0" ::: "memory");
    }
    __syncthreads();

    const float4* __restrict__ x4 = (const float4*)x;
    float4* __restrict__ o4 = (float4*)out;
    int stride = (int)(gridDim.x * blockDim.x);
    int gtid = (int)(blockIdx.x * blockDim.x + threadIdx.x);

    for (int i = gtid; i < n4; i += stride) {
        if (i + stride < n4) __builtin_prefetch(x4 + i + stride, 0, 1);
        float4 a = x4[i];
        float y0, y1, y2, y3, t0, t1, t2, t3;
        eval_core(a.x, sC, y0, t0);
        eval_core(a.y, sC, y1, t1);
        eval_core(a.z, sC, y2, t2);
        eval_core(a.w, sC, y3, t3);
        bool f0 = fabsf(a.x) > 6.0f, f1 = fabsf(a.y) > 6.0f,
             f2 = fabsf(a.z) > 6.0f, f3 = fabsf(a.w) > 6.0f;
        // |x|>6 is ~6e-5 of lanes: whole waves skip the log path (wave32 ballot).
        // Region C (x>=6.25) is a subset of this guard, so the hot path never
        // evaluates the C rational.
        if (__builtin_amdgcn_ballot_w32(f0 | f1 | f2 | f3)) {
            if (f0) { y0 = asym_far(fabsf(a.x)); if (a.x >= 6.25f) t0 = temp_far_C(a.x); }
            if (f1) { y1 = asym_far(fabsf(a.y)); if (a.y >= 6.25f) t1 = temp_far_C(a.y); }
            if (f2) { y2 = asym_far(fabsf(a.z)); if (a.z >= 6.25f) t2 = temp_far_C(a.z); }
            if (f3) { y3 = asym_far(fabsf(a.w)); if (a.w >= 6.25f) t3 = temp_far_C(a.w); }
        }
        float4 r;
        r.x = y0 + t0; r.y = y1 + t1; r.z = y2 + t2; r.w = y3 + t3;
        o4[i] = r;
    }

    // tail (n % 4 elements)
    if (gtid < rem) {
        int i = n4 * 4 + gtid;
        float xi = x[i];
        float y, tm;
        eval_core(xi, sC, y, tm);
        float ax = fabsf(xi);
        if (ax > 6.0f) {
            y = asym_far(ax);
            if (xi >= 6.25f) tm = temp_far_C(xi);
        }
        out[i] = y + tm;
    }
}

extern "C" void kernel_launch(void* const* d_in, const int* in_sizes, int n_in,
                              void* d_out, int out_size, void* d_ws, size_t ws_size,
                              hipStream_t stream) {
    (void)n_in; (void)out_size; (void)d_ws; (void)ws_size;
    const float* x    = (const float*)d_in[0];
    const float* cheb = (const float*)d_in[1];
    float* out = (float*)d_out;

    int n   = in_sizes[0];
    int n4  = n >> 2;
    int rem = n & 3;

    int blocks = (n4 + BLOCK - 1) / BLOCK;
    if (blocks > 4096) blocks = 4096;   // grid-stride; keeps prefetch meaningful
    if (blocks < 1) blocks = 1;
    dawson_g_kernel<<<blocks, BLOCK, 0, stream>>>(x, cheb, out, n4, rem);
}